// docHInAttnM1_38886633898203
// MI455X (gfx1250) — compile-verified
//
#include <hip/hip_runtime.h>

// Problem dims (match reference)
#define B_ 8192
#define L_ 64
#define D_ 200

#ifndef __has_builtin
#define __has_builtin(x) 0
#endif

// CDNA5 async global->LDS copy (ASYNCcnt-tracked). Guarded so host-side
// compilation (and any toolchain lacking the builtin) falls back to a plain
// staged copy.
#if __has_builtin(__builtin_amdgcn_global_load_async_to_lds_b128)
#define ASYNC_LDS 1
#else
#define ASYNC_LDS 0
#endif

#if ASYNC_LDS
typedef __attribute__((ext_vector_type(4))) int v4i_;
typedef __attribute__((address_space(1))) v4i_* gptr_v4i;
typedef __attribute__((address_space(3))) v4i_* lptr_v4i;

// Issue K async 16B/lane transfers with constant immediate offsets
// (INST_OFFSET is added to BOTH the global and the LDS address).
// Each step advances by 256 lanes * 16B = 4096 bytes.
template <int K>
struct AsyncCopyUnroll {
    static __device__ __forceinline__ void run(gptr_v4i g, lptr_v4i l) {
        AsyncCopyUnroll<K - 1>::run(g, l);
        __builtin_amdgcn_global_load_async_to_lds_b128(
            g, l, /*offset=*/(K - 1) * 4096, /*cpol=*/0);
    }
};
template <>
struct AsyncCopyUnroll<0> {
    static __device__ __forceinline__ void run(gptr_v4i, lptr_v4i) {}
};
#endif

// Hardware tanh when available (V_TANH_F32, single trans op). Fallback:
// one v_exp_f32 based formula; absolute error ~1 ulp of 1.0, fine for
// softmax scores. (libm tanhf is a ~25-op polynomial and would make this
// kernel VALU-bound instead of HBM-bound.)
__device__ __forceinline__ float fast_tanh(float x) {
#if __has_builtin(__builtin_amdgcn_tanhf)
    return __builtin_amdgcn_tanhf(x);
#else
    float e = __expf(2.f * x);
    return 1.f - 2.f / (e + 1.f);
#endif
}

__global__ __launch_bounds__(256) void attn_pool_kernel(
    const float* __restrict__ V,        // [B, L, D]
    const int* __restrict__ lengths,    // [B]
    const float* __restrict__ w,        // [D]
    const float* __restrict__ dW1,      // scalar
    float* __restrict__ out)            // [B, D]
{
    __shared__ float Vlds[L_ * D_];     // 51200 B: one sentence tile
    __shared__ float sc[L_];            // scores, then alpha
    __shared__ float wl[D_];            // staged weight vector

    const int b   = blockIdx.x;
    const int tid = threadIdx.x;
    const float dw = dW1[0];            // uniform scalar load

    // ---- Stage V[b] (64x200 f32, contiguous, 16B-aligned) into LDS ----
    // 3200 float4 transfers = 256 lanes * 12 full steps + 128-lane tail.
    const float4* gsrc = (const float4*)(V + (size_t)b * (L_ * D_));
#if ASYNC_LDS
    {
        gptr_v4i g = (gptr_v4i)(gsrc + tid);
        lptr_v4i l = (lptr_v4i)(&Vlds[tid * 4]);
        AsyncCopyUnroll<12>::run(g, l);          // offsets 0 .. 11*4096
        if (tid < 128)
            __builtin_amdgcn_global_load_async_to_lds_b128(
                g, l, /*offset=*/12 * 4096, /*cpol=*/0);
    }
#else
    for (int i = tid; i < (L_ * D_) / 4; i += 256)
        ((float4*)Vlds)[i] = gsrc[i];
#endif
    if (tid < D_) wl[tid] = w[tid];

#if ASYNC_LDS
#if __has_builtin(__builtin_amdgcn_s_wait_asynccnt)
    __builtin_amdgcn_s_wait_asynccnt(0);
#else
    asm volatile("s_wait_asynccnt 0" ::: "memory");
#endif
#endif
    __syncthreads();

    // ---- Pass 1: scores[l] = sum_d tanh(dw * V[l,d]) * w[d] ----
    // 4 lanes per token; d strided by 4 -> conflict-free LDS banks
    // (200 mod 64 == 8, so banks 8l+p are distinct across a wave).
    {
        const int l = tid >> 2;
        const int p = tid & 3;
        const float* row = &Vlds[l * D_];
        float acc = 0.f;
#pragma unroll 10
        for (int d = p; d < D_; d += 4)
            acc = fmaf(fast_tanh(dw * row[d]), wl[d], acc);
        // reduce the group of 4 adjacent lanes (same wave32)
        acc += __shfl_xor(acc, 1, 32);
        acc += __shfl_xor(acc, 2, 32);
        if (p == 0) sc[l] = acc;
    }
    __syncthreads();

    // ---- Pass 2: masked softmax over 64 tokens, done by wave 0 ----
    if (tid < 32) {
        int len = lengths[b];
        len = len < 1 ? 1 : (len > L_ ? L_ : len);   // clip(lengths, 1, L)
        const float NEG_INF = -__builtin_inff();
        float s0 = (tid        < len) ? sc[tid]      : NEG_INF;
        float s1 = ((tid + 32) < len) ? sc[tid + 32] : NEG_INF;

        float m = fmaxf(s0, s1);
        for (int off = 16; off >= 1; off >>= 1)
            m = fmaxf(m, __shfl_xor(m, off, 32));

        float e0 = expf(s0 - m);   // exp(-inf) == 0 for masked tokens
        float e1 = expf(s1 - m);
        float sum = e0 + e1;
        for (int off = 16; off >= 1; off >>= 1)
            sum += __shfl_xor(sum, off, 32);

        const float inv = 1.f / sum;
        sc[tid]      = e0 * inv;
        sc[tid + 32] = e1 * inv;
    }
    __syncthreads();

    // ---- Pass 3: out[d] = sum_l alpha[l] * V[l,d] ----
    // Consecutive addresses across lanes -> conflict-free; sc[l] broadcasts.
    if (tid < D_) {
        float acc = 0.f;
#pragma unroll 8
        for (int l = 0; l < L_; ++l)
            acc = fmaf(sc[l], Vlds[l * D_ + tid], acc);
        out[(size_t)b * D_ + tid] = acc;
    }
}

extern "C" void kernel_launch(void* const* d_in, const int* in_sizes, int n_in,
                              void* d_out, int out_size, void* d_ws, size_t ws_size,
                              hipStream_t stream) {
    const float* V       = (const float*)d_in[0];
    const int*   lengths = (const int*)  d_in[1];
    const float* w       = (const float*)d_in[2];
    const float* dW1     = (const float*)d_in[3];
    float* out = (float*)d_out;

    (void)in_sizes; (void)n_in; (void)out_size; (void)d_ws; (void)ws_size;

    attn_pool_kernel<<<B_, 256, 0, stream>>>(V, lengths, w, dW1, out);
}